// ModelNew_80908593923007
// MI455X (gfx1250) — compile-verified
//
#include <hip/hip_runtime.h>

typedef float v2f __attribute__((ext_vector_type(2)));
typedef float v8f __attribute__((ext_vector_type(8)));

// Problem constants
#define CIN   8
#define COUT  32
#define DD    16
#define HH    128
#define WW    128
#define KTOT  216        // 8*3*3*3
#define NPOS  504        // 2*2*126 conv outputs per block
#define NPAD  512
#define DZB   7          // pooled d windows
#define HYB   63         // pooled h windows
#define NBLK_PER_B (DZB*HYB)   // 441
#define POOLDIV 27783.0f       // 7*63*63

// smem layout (floats): [0,16384) xtile / result reuse ; [16384, 16384+6912) weights
#define XT_SZ  16384
#define W_OFF  16384
#define SMEM_F (XT_SZ + COUT*KTOT)   // 23296 floats = 93184 B

__global__ __launch_bounds__(256)
void conv3d_pool_wmma_kernel(const float* __restrict__ x,
                             const float* __restrict__ conv_weight,
                             const float* __restrict__ conv_bias,
                             float* __restrict__ partials)
{
    __shared__ __align__(16) float smem[SMEM_F];
    __shared__ int   base_off[KTOT];
    __shared__ float cbias[COUT];
    __shared__ float red[8];

    const int tid  = threadIdx.x;
    const int hy   = blockIdx.x;   // 0..62
    const int dz   = blockIdx.y;   // 0..6
    const int b    = blockIdx.z;   // 0..15

    // ---- Phase 0: cooperative loads -------------------------------------
    // weights: [32][216] contiguous as given (OIDHW flattens to [cout][k])
    for (int i = tid; i < COUT * KTOT; i += 256)
        smem[W_OFF + i] = conv_weight[i];
    if (tid < COUT) cbias[tid] = conv_bias[tid];
    if (tid < KTOT) {
        int ci = tid / 27;
        int r  = tid - ci * 27;
        int kd = r / 9;
        int r2 = r - kd * 9;
        int kh = r2 / 3;
        int kw = r2 - kh * 3;
        base_off[tid] = ((ci * 4 + kd) * 4 + kh) * 128 + kw;
    }
    // x tile: [ci:8][d4:4][h4:4][w:128] = 128 rows x 128 floats
    {
        const int row  = tid >> 1;        // 0..127
        const int half = tid & 1;
        const int ci = row >> 4;
        const int d4 = (row >> 2) & 3;
        const int h4 = row & 3;
        const float* src = x
            + ((size_t)((b * CIN + ci) * DD + (2 * dz + d4)) * HH + (2 * hy + h4)) * WW
            + half * 64;
        float* dst = smem + row * 128 + half * 64;
        const float4* s4 = (const float4*)src;
        float4*       d4p = (float4*)dst;
#pragma unroll
        for (int i = 0; i < 16; ++i) d4p[i] = s4[i];
    }
    __syncthreads();

    // ---- Phase 1: implicit GEMM via V_WMMA_F32_16X16X4_F32 --------------
    const int lane = tid & 31;
    const int wave = tid >> 5;
    const int m    = lane & 15;
    const int hf   = lane >> 4;

    // per-lane column offsets for the 4 N-tiles this wave owns
    int noff[4];
#pragma unroll
    for (int t = 0; t < 4; ++t) {
        int n = (t * 8 + wave) * 16 + m;     // global position 0..511
        if (n < NPOS) {
            int dd  = n / 252;
            int rem = n - dd * 252;
            int h2  = rem / 126;
            int ww  = rem - h2 * 126;
            noff[t] = dd * 512 + h2 * 128 + ww;
        } else {
            noff[t] = 0;   // padded columns compute garbage, never consumed
        }
    }

    v8f acc[2][4] = {};
    const float* Wl = smem + W_OFF;
#pragma unroll 2
    for (int k0 = 0; k0 < KTOT; k0 += 4) {
        const int ka = k0 + 2 * hf;                 // frag elems: K = ka, ka+1
        v2f a0 = *(const v2f*)(Wl + m * KTOT + ka);          // cout 0..15
        v2f a1 = *(const v2f*)(Wl + (m + 16) * KTOT + ka);   // cout 16..31
        const int bo0 = base_off[ka];
        const int bo1 = base_off[ka + 1];
#pragma unroll
        for (int t = 0; t < 4; ++t) {
            v2f bt;
            bt.x = smem[bo0 + noff[t]];
            bt.y = smem[bo1 + noff[t]];
            acc[0][t] = __builtin_amdgcn_wmma_f32_16x16x4_f32(
                false, a0, false, bt, (short)0, acc[0][t], false, false);
            acc[1][t] = __builtin_amdgcn_wmma_f32_16x16x4_f32(
                false, a1, false, bt, (short)0, acc[1][t], false, false);
        }
    }

    // ---- Phase 2: epilogue (bias, /2) -> result LDS (reuse x tile) ------
    __syncthreads();   // everyone done reading xtile
#pragma unroll
    for (int mt = 0; mt < 2; ++mt) {
#pragma unroll
        for (int t = 0; t < 4; ++t) {
            const int nl = (t * 8 + wave) * 16 + m;
#pragma unroll
            for (int e = 0; e < 8; ++e) {
                const int c = mt * 16 + hf * 8 + e;       // C/D layout: M = e + 8*half
                smem[c * NPAD + nl] = (acc[mt][t][e] + cbias[c]) * 0.5f;
            }
        }
    }
    __syncthreads();

    // ---- Phase 3: 2x2x2 max-pool + block sum ----------------------------
    float s = 0.0f;
    for (int idx = tid; idx < COUT * HYB; idx += 256) {
        const int c  = idx / HYB;
        const int wz = idx - c * HYB;
        const float* base = smem + c * NPAD + 2 * wz;
        float mv = base[0];
#pragma unroll
        for (int dd = 0; dd < 2; ++dd)
#pragma unroll
            for (int h2 = 0; h2 < 2; ++h2)
#pragma unroll
                for (int dw = 0; dw < 2; ++dw)
                    mv = fmaxf(mv, base[dd * 252 + h2 * 126 + dw]);
        s += mv;
    }
#pragma unroll
    for (int off = 16; off > 0; off >>= 1)
        s += __shfl_xor(s, off, 32);
    if (lane == 0) red[wave] = s;
    __syncthreads();
    if (tid == 0) {
        float tot = 0.0f;
#pragma unroll
        for (int w = 0; w < 8; ++w) tot += red[w];
        partials[(b * DZB + dz) * HYB + hy] = tot;
    }
}

__global__ __launch_bounds__(128)
void finalize_kernel(const float* __restrict__ partials,
                     const float* __restrict__ bias,
                     float* __restrict__ out)
{
    __shared__ float r[128];
    const int b = blockIdx.x;
    float s = 0.0f;
    for (int i = threadIdx.x; i < NBLK_PER_B; i += 128)
        s += partials[b * NBLK_PER_B + i];
    r[threadIdx.x] = s;
    __syncthreads();
    for (int st = 64; st > 0; st >>= 1) {
        if (threadIdx.x < st) r[threadIdx.x] += r[threadIdx.x + st];
        __syncthreads();
    }
    if (threadIdx.x == 0) {
        float bs = 0.0f;
        for (int c = 0; c < COUT; ++c) bs += bias[c];
        out[b] = r[0] * (1.0f / POOLDIV) + bs;
    }
}

extern "C" void kernel_launch(void* const* d_in, const int* in_sizes, int n_in,
                              void* d_out, int out_size, void* d_ws, size_t ws_size,
                              hipStream_t stream)
{
    const float* x           = (const float*)d_in[0];
    const float* conv_weight = (const float*)d_in[1];
    const float* conv_bias   = (const float*)d_in[2];
    const float* bias        = (const float*)d_in[3];
    float* out      = (float*)d_out;
    float* partials = (float*)d_ws;   // 16*441 floats = 28224 B

    dim3 grid(HYB, DZB, 16);
    conv3d_pool_wmma_kernel<<<grid, 256, 0, stream>>>(x, conv_weight, conv_bias, partials);
    finalize_kernel<<<16, 128, 0, stream>>>(partials, bias, out);
}